// SparseAutoencoder_80427557585146
// MI455X (gfx1250) — compile-verified
//
#include <hip/hip_runtime.h>

#define D_MODEL   768
#define D_SPARSE  24576
#define N_TOKENS  4096
#define TOPK      32

typedef __attribute__((ext_vector_type(2))) float v2f;
typedef __attribute__((ext_vector_type(8))) float v8f;

// ---------------------------------------------------------------------------
// Kernel 1: encode GEMM via V_WMMA_F32_16X16X4_F32
//   hidden[n][s] = relu( (x[n]-dec_bias) . W_enc[s] + enc_bias[s] )
// Block = 256 thr (8 waves). Block tile 128(M) x 256(N); wave tile 64x64
// (4x4 WMMA tiles, 128 acc VGPRs) -> 16 WMMAs per 8 fragment loads per K-step,
// halving operand-delivery bandwidth per matrix op vs a 4x2 tile.
// A layout: lane<16 holds K=k,k+1 ; lane>=16 holds K=k+2,k+3.
// ---------------------------------------------------------------------------
__global__ __launch_bounds__(256) void sae_encode_wmma(
    const float* __restrict__ X, const float* __restrict__ Wenc,
    const float* __restrict__ enc_bias, const float* __restrict__ dec_bias,
    float* __restrict__ hidden)
{
  const int lane = threadIdx.x & 31;
  const int wave = threadIdx.x >> 5;
  const int half = lane >> 4;     // 0: K pair 0, 1: K pair 1
  const int l    = lane & 15;     // row (A) / col (B) within tile
  const int wm   = wave >> 2;     // 0..1
  const int wn   = wave & 3;      // 0..3
  const int m0   = blockIdx.y * 128 + wm * 64;
  const int n0   = blockIdx.x * 256 + wn * 64;

  v8f acc[4][4];
#pragma unroll
  for (int i = 0; i < 4; ++i)
#pragma unroll
    for (int j = 0; j < 4; ++j)
      acc[i][j] = (v8f){0.f,0.f,0.f,0.f,0.f,0.f,0.f,0.f};

  const float* xr[4];
  const float* wr[4];
#pragma unroll
  for (int i = 0; i < 4; ++i) xr[i] = X    + (size_t)(m0 + i * 16 + l) * D_MODEL;
#pragma unroll
  for (int j = 0; j < 4; ++j) wr[j] = Wenc + (size_t)(n0 + j * 16 + l) * D_MODEL;

  for (int k = 0; k < D_MODEL; k += 4) {
    const int kk = k + 2 * half;
    const v2f db = *(const v2f*)(dec_bias + kk);
    v2f a[4], b[4];
#pragma unroll
    for (int i = 0; i < 4; ++i) a[i] = *(const v2f*)(xr[i] + kk) - db;
#pragma unroll
    for (int j = 0; j < 4; ++j) b[j] = *(const v2f*)(wr[j] + kk);
#pragma unroll
    for (int i = 0; i < 4; ++i)
#pragma unroll
      for (int j = 0; j < 4; ++j)
        acc[i][j] = __builtin_amdgcn_wmma_f32_16x16x4_f32(
            false, a[i], false, b[j], (short)0, acc[i][j], false, false);
  }

  // Store with relu + enc_bias. C/D layout: VGPR r, lanes 0-15 -> row r,
  // lanes 16-31 -> row r+8; column = lane%16. Stream out (non-temporal).
#pragma unroll
  for (int j = 0; j < 4; ++j) {
    const int col = n0 + j * 16 + l;
    const float eb = enc_bias[col];
#pragma unroll
    for (int i = 0; i < 4; ++i) {
      const int row0 = m0 + i * 16 + 8 * half;
#pragma unroll
      for (int r = 0; r < 8; ++r) {
        float v = acc[i][j][r] + eb;
        v = fmaxf(v, 0.0f);
        __builtin_nontemporal_store(v, hidden + (size_t)(row0 + r) * D_SPARSE + col);
      }
    }
  }
}

// ---------------------------------------------------------------------------
// Kernel 2: W_dec [768, 24576] -> W_decT [24576, 768] (coalesced LDS transpose)
// ---------------------------------------------------------------------------
__global__ __launch_bounds__(256) void wdec_transpose(
    const float* __restrict__ Wdec, float* __restrict__ WdecT)
{
  __shared__ float tile[32][33];
  const int tx = threadIdx.x;             // 0..31
  const int ty = threadIdx.y;             // 0..7
  const int x0 = blockIdx.x * 32;         // s
  const int y0 = blockIdx.y * 32;         // d
#pragma unroll
  for (int i = 0; i < 4; ++i)
    tile[ty + 8 * i][tx] = Wdec[(size_t)(y0 + ty + 8 * i) * D_SPARSE + (x0 + tx)];
  __syncthreads();
#pragma unroll
  for (int i = 0; i < 4; ++i)
    WdecT[(size_t)(x0 + ty + 8 * i) * D_MODEL + (y0 + tx)] = tile[tx][ty + 8 * i];
}

// ---------------------------------------------------------------------------
// Kernel 3: per-token radix top-k select (k=32 over 24576 non-negative f32).
// 4 histogram passes over the float bit pattern (monotone for >=0 floats)
// find the exact k-th threshold; then keep >T plus `remaining` ties, zero the
// rest in-place, and emit a compact (idx,val) list for decode.
// One 256-thread block per token; row stays L2-resident across passes.
// ---------------------------------------------------------------------------
__global__ __launch_bounds__(256) void sae_topk(
    float* __restrict__ hidden, int* __restrict__ idxL, float* __restrict__ valL)
{
  __shared__ unsigned int hist[256];
  __shared__ unsigned int s_sel, s_rem;
  __shared__ unsigned int tieClaim, outPos;
  const int tid = threadIdx.x;
  const size_t n = blockIdx.x;
  float* row = hidden + n * (size_t)D_SPARSE;

  unsigned int prefix = 0, done_mask = 0, remaining = TOPK;

  for (int shift = 24; shift >= 0; shift -= 8) {
    hist[tid] = 0;
    __syncthreads();
    for (int i = tid; i < D_SPARSE; i += 256) {
      const unsigned int key = __float_as_uint(row[i]);
      if (((key ^ prefix) & done_mask) == 0)
        atomicAdd(&hist[(key >> shift) & 255], 1u);
    }
    __syncthreads();
    if (tid == 0) {
      unsigned int cum = 0;
      int b = 255;
      for (; b > 0; --b) {
        const unsigned int c = hist[b];
        if (cum + c >= remaining) break;
        cum += c;
      }
      s_sel = (unsigned int)b;
      s_rem = remaining - cum;
    }
    __syncthreads();
    prefix |= s_sel << shift;
    remaining = s_rem;
    done_mask |= 0xFFu << shift;
    __syncthreads();
  }

  if (tid == 0) { tieClaim = 0; outPos = 0; }
  __syncthreads();

  const unsigned int T = prefix;
  for (int i = tid; i < D_SPARSE; i += 256) {
    const float v = row[i];
    const unsigned int key = __float_as_uint(v);
    bool keep;
    if (key > T)       keep = true;
    else if (key == T) keep = (atomicAdd(&tieClaim, 1u) < remaining);
    else               keep = false;
    if (keep) {
      const unsigned int p = atomicAdd(&outPos, 1u);
      idxL[n * TOPK + p] = i;
      valL[n * TOPK + p] = v;
    } else if (key != 0u) {
      __builtin_nontemporal_store(0.0f, row + i);   // zero non-kept (skip already-zero)
    }
  }
}

// ---------------------------------------------------------------------------
// Kernel 4: sparse decode + error. One block per token; thread d covers
// d, d+256, d+512. Coalesced gathers of 32 W_decT rows (L2 resident).
// Deterministic block tree-reduction of sum(e^2) into partial[n].
// ---------------------------------------------------------------------------
__global__ __launch_bounds__(256) void sae_decode(
    const float* __restrict__ WdecT, const float* __restrict__ dec_bias,
    const float* __restrict__ X, const int* __restrict__ idxL,
    const float* __restrict__ valL, float* __restrict__ sae_out,
    float* __restrict__ partial)
{
  __shared__ float s_val[TOPK];
  __shared__ int   s_idx[TOPK];
  __shared__ float red[256];
  const int tid = threadIdx.x;
  const size_t n = blockIdx.x;

  if (tid < TOPK) {
    s_idx[tid] = idxL[n * TOPK + tid];
    s_val[tid] = valL[n * TOPK + tid];
  }
  __syncthreads();

  float acc0 = dec_bias[tid];
  float acc1 = dec_bias[tid + 256];
  float acc2 = dec_bias[tid + 512];
#pragma unroll 4
  for (int j = 0; j < TOPK; ++j) {
    const float* w = WdecT + (size_t)s_idx[j] * D_MODEL;
    const float vj = s_val[j];
    acc0 = fmaf(vj, w[tid],       acc0);
    acc1 = fmaf(vj, w[tid + 256], acc1);
    acc2 = fmaf(vj, w[tid + 512], acc2);
  }

  const float* xr  = X       + n * (size_t)D_MODEL;
  float*       orw = sae_out + n * (size_t)D_MODEL;
  float es = 0.f, e;
  orw[tid]       = acc0; e = acc0 - xr[tid];       es = fmaf(e, e, es);
  orw[tid + 256] = acc1; e = acc1 - xr[tid + 256]; es = fmaf(e, e, es);
  orw[tid + 512] = acc2; e = acc2 - xr[tid + 512]; es = fmaf(e, e, es);

  red[tid] = es;
  __syncthreads();
  for (int s = 128; s > 0; s >>= 1) {
    if (tid < s) red[tid] += red[tid + s];
    __syncthreads();
  }
  if (tid == 0) partial[n] = red[0];
}

// ---------------------------------------------------------------------------
// Kernel 5: deterministic final reduction -> l2_loss, reconstruction_loss
// ---------------------------------------------------------------------------
__global__ __launch_bounds__(256) void sae_finalize(
    const float* __restrict__ partial, float* __restrict__ loss)
{
  __shared__ float red[256];
  const int tid = threadIdx.x;
  float s = 0.f;
  for (int i = tid; i < N_TOKENS; i += 256) s += partial[i];
  red[tid] = s;
  __syncthreads();
  for (int k = 128; k > 0; k >>= 1) {
    if (tid < k) red[tid] += red[tid + k];
    __syncthreads();
  }
  if (tid == 0) {
    const float l2 = red[0];
    loss[0] = l2;
    loss[1] = l2 / (float)((size_t)N_TOKENS * D_MODEL);
  }
}

// ---------------------------------------------------------------------------
extern "C" void kernel_launch(void* const* d_in, const int* in_sizes, int n_in,
                              void* d_out, int out_size, void* d_ws, size_t ws_size,
                              hipStream_t stream)
{
  (void)in_sizes; (void)n_in; (void)out_size; (void)ws_size;
  const float* x        = (const float*)d_in[0];
  const float* Wenc     = (const float*)d_in[1];
  const float* enc_bias = (const float*)d_in[2];
  const float* Wdec     = (const float*)d_in[3];
  const float* dec_bias = (const float*)d_in[4];
  // d_in[5] = top_k (compile-time TOPK=32)

  float* out     = (float*)d_out;
  float* sae_out = out;
  float* hidden  = out + (size_t)N_TOKENS * D_MODEL;
  float* loss    = hidden + (size_t)N_TOKENS * D_SPARSE;  // [l2, recon]

  char*  ws      = (char*)d_ws;
  float* WdecT   = (float*)ws;                                         // 24576*768 f32
  int*   idxL    = (int*)(ws + sizeof(float) * (size_t)D_SPARSE * D_MODEL);
  float* valL    = (float*)(idxL + (size_t)N_TOKENS * TOPK);
  float* partial = valL + (size_t)N_TOKENS * TOPK;

  hipLaunchKernelGGL(wdec_transpose, dim3(D_SPARSE / 32, D_MODEL / 32), dim3(32, 8),
                     0, stream, Wdec, WdecT);
  hipLaunchKernelGGL(sae_encode_wmma, dim3(D_SPARSE / 256, N_TOKENS / 128), dim3(256),
                     0, stream, x, Wenc, enc_bias, dec_bias, hidden);
  hipLaunchKernelGGL(sae_topk, dim3(N_TOKENS), dim3(256),
                     0, stream, hidden, idxL, valL);
  hipLaunchKernelGGL(sae_decode, dim3(N_TOKENS), dim3(256),
                     0, stream, WdecT, dec_bias, x, idxL, valL, sae_out, partial);
  hipLaunchKernelGGL(sae_finalize, dim3(1), dim3(256),
                     0, stream, partial, loss);
}